// LightGCN_68985764708540
// MI455X (gfx1250) — compile-verified
//
#include <hip/hip_runtime.h>
#include <stdint.h>

#define N_NODES   100000
#define EMBED     64
#define N_EDGES   3200000
#define N_LAYERS  3

#define CH        32            // CSR entries staged per chunk (1 x b64 per lane)
#define WAVES     8             // nodes (waves) per block
#define TPB       (WAVES * 32)  // wave32

// ---------------- CDNA5 async global->LDS (ASYNCcnt path), guarded ----------------
#if defined(__HIP_DEVICE_COMPILE__) && defined(__gfx1250__)
#define ASYNC_OK __has_builtin(__builtin_amdgcn_global_load_async_to_lds_b64)
#else
#define ASYNC_OK 0
#endif

#if ASYNC_OK
typedef int v2i __attribute__((vector_size(8)));                  // 2 x i32 (8B)
typedef __attribute__((address_space(1))) v2i* g_v2i_ptr;         // global
typedef __attribute__((address_space(3))) v2i* l_v2i_ptr;         // LDS

__device__ __forceinline__ void stage_b64(const void* g, void* l) {
  // (global src, lds dst, imm offset, imm cpol)
  __builtin_amdgcn_global_load_async_to_lds_b64(
      (g_v2i_ptr)(uintptr_t)g,
      (l_v2i_ptr)(uint32_t)(uintptr_t)l,
      0, 0);
}
#if __has_builtin(__builtin_amdgcn_s_wait_asynccnt)
#define WAIT_ASYNC(n) do { __builtin_amdgcn_s_wait_asynccnt(n); asm volatile("" ::: "memory"); } while (0)
#else
#define WAIT_ASYNC(n) asm volatile("s_wait_asynccnt %0" :: "i"(n) : "memory")
#endif
#else
__device__ __forceinline__ void stage_b64(const void* g, void* l) {
  *(unsigned long long*)l = *(const unsigned long long*)g;  // sync fallback
}
#define WAIT_ASYNC(n) do { } while (0)
#endif

// ---------------- Kernels ----------------

// In-degree histogram over edge targets (col).
__global__ void k_degree(const long long* __restrict__ col, unsigned* __restrict__ deg) {
  int e = blockIdx.x * blockDim.x + threadIdx.x;
  if (e < N_EDGES) atomicAdd(&deg[(int)col[e]], 1u);
}

// dinv[i] = deg>0 ? rsqrt(deg) : 0
__global__ void k_dinv(const unsigned* __restrict__ deg, float* __restrict__ dinv) {
  int i = blockIdx.x * blockDim.x + threadIdx.x;
  if (i < N_NODES) {
    unsigned d = deg[i];
    dinv[i] = d ? rsqrtf((float)d) : 0.0f;
  }
}

// Single-workgroup exclusive scan of deg -> CSR row offsets (tiny: 100K u32).
__global__ void k_scan(const unsigned* __restrict__ deg, unsigned* __restrict__ offs,
                       unsigned* __restrict__ cursor) {
  __shared__ unsigned sh[1024];
  __shared__ unsigned carry;
  const int tid = threadIdx.x;
  if (tid == 0) carry = 0u;
  __syncthreads();
  for (int base = 0; base < N_NODES; base += 1024) {
    int i = base + tid;
    unsigned v = (i < N_NODES) ? deg[i] : 0u;
    sh[tid] = v;
    __syncthreads();
    for (int off = 1; off < 1024; off <<= 1) {
      unsigned t = (tid >= off) ? sh[tid - off] : 0u;
      __syncthreads();
      sh[tid] += t;
      __syncthreads();
    }
    unsigned excl = sh[tid] - v + carry;
    if (i < N_NODES) { offs[i] = excl; cursor[i] = excl; }
    __syncthreads();
    if (tid == 0) carry += sh[1023];
    __syncthreads();
  }
  if (tid == 0) offs[N_NODES] = carry;
}

// Place each edge into its target's CSR segment: entry = {src, norm} packed in 8B.
__global__ void k_fill(const long long* __restrict__ row, const long long* __restrict__ col,
                       const float* __restrict__ dinv, unsigned* __restrict__ cursor,
                       int2* __restrict__ csr) {
  int e = blockIdx.x * blockDim.x + threadIdx.x;
  if (e >= N_EDGES) return;
  int r = (int)row[e];
  int c = (int)col[e];
  unsigned p = atomicAdd(&cursor[c], 1u);
  int2 ent;
  ent.x = r;
  ent.y = __float_as_int(dinv[r] * dinv[c]);
  csr[p] = ent;
}

// x_cur = emb ; acc = emb / 4
__global__ void k_init(const float* __restrict__ emb, float* __restrict__ xc,
                       float* __restrict__ acc) {
  int t = blockIdx.x * blockDim.x + threadIdx.x;
  if (t < N_NODES * EMBED) {
    float v = emb[t];
    xc[t] = v;
    acc[t] = 0.25f * v;
  }
}

// One LGConv layer: out[i] = sum_{(j,i)} norm_ji * x[j]; acc += out/4.
// One wave per node; lane owns a float2 feature pair (64 f32 = 32 lanes x float2).
// CSR entries double-buffered into LDS via async global->LDS (ASYNCcnt).
__global__ __launch_bounds__(TPB) void k_prop(const float2* __restrict__ xc,
                                              float2* __restrict__ xn,
                                              float2* __restrict__ acc,
                                              const unsigned* __restrict__ offs,
                                              const int2* __restrict__ csr) {
  __shared__ int2 stage[WAVES][2][CH];
  const int wv   = threadIdx.x >> 5;
  const int lane = threadIdx.x & 31;
  const int node = blockIdx.x * WAVES + wv;
  if (node >= N_NODES) return;

  const unsigned s0 = offs[node];
  const unsigned s1 = offs[node + 1];
  const int cnt = (int)(s1 - s0);
  const int nc  = (cnt + CH - 1) / CH;

  if (nc > 0) {
    unsigned idx = s0 + (unsigned)lane;
    if (idx < s1) stage_b64(&csr[idx], &stage[wv][0][lane]);
  }

  float2 sum = make_float2(0.0f, 0.0f);
  for (int c = 0; c < nc; ++c) {
    const int buf = c & 1;
    if (c + 1 < nc) {
      unsigned idx = s0 + (unsigned)((c + 1) * CH + lane);
      if (idx < s1) stage_b64(&csr[idx], &stage[wv][buf ^ 1][lane]);
      WAIT_ASYNC(1);  // chunk c complete (in-order), c+1 still in flight
    } else {
      WAIT_ASYNC(0);
    }
    const int kend = (cnt - c * CH < CH) ? (cnt - c * CH) : CH;
#pragma unroll 4
    for (int k = 0; k < kend; ++k) {
      int2 ent  = stage[wv][buf][k];              // LDS broadcast read
      float w   = __int_as_float(ent.y);
      float2 xv = xc[(size_t)ent.x * 32 + lane];  // 256B coalesced row gather
      sum.x = fmaf(w, xv.x, sum.x);
      sum.y = fmaf(w, xv.y, sum.y);
    }
  }

  const size_t o = (size_t)node * 32 + lane;
  xn[o] = sum;
  float2 a = acc[o];
  a.x = fmaf(0.25f, sum.x, a.x);
  a.y = fmaf(0.25f, sum.y, a.y);
  acc[o] = a;
}

// ---------------- Host launcher ----------------
extern "C" void kernel_launch(void* const* d_in, const int* in_sizes, int n_in,
                              void* d_out, int out_size, void* d_ws, size_t ws_size,
                              hipStream_t stream) {
  (void)in_sizes; (void)n_in; (void)out_size; (void)ws_size;

  const float*     emb = (const float*)d_in[0];
  const long long* ei  = (const long long*)d_in[1];  // int64 [2, E]
  const long long* row = ei;
  const long long* col = ei + N_EDGES;
  float* acc = (float*)d_out;

  // Workspace carve-up (256B-aligned slices), total ~79 MB.
  char*  ws  = (char*)d_ws;
  size_t off = 0;
  auto carve = [&](size_t bytes) -> char* {
    char* p = ws + off;
    off += (bytes + 255) & ~(size_t)255;
    return p;
  };
  unsigned* deg    = (unsigned*)carve((size_t)N_NODES * 4);
  float*    dinv   = (float*)   carve((size_t)N_NODES * 4);
  unsigned* offs   = (unsigned*)carve((size_t)(N_NODES + 1) * 4);
  unsigned* cursor = (unsigned*)carve((size_t)N_NODES * 4);
  int2*     csr    = (int2*)    carve((size_t)N_EDGES * 8);
  float*    x_a    = (float*)   carve((size_t)N_NODES * EMBED * 4);
  float*    x_b    = (float*)   carve((size_t)N_NODES * EMBED * 4);

  (void)hipMemsetAsync(deg, 0, (size_t)N_NODES * 4, stream);

  const int eb = (N_EDGES + 255) / 256;
  const int nb = (N_NODES + 255) / 256;

  k_degree<<<eb, 256, 0, stream>>>(col, deg);
  k_dinv  <<<nb, 256, 0, stream>>>(deg, dinv);
  k_scan  <<<1, 1024, 0, stream>>>(deg, offs, cursor);
  k_fill  <<<eb, 256, 0, stream>>>(row, col, dinv, cursor, csr);
  k_init  <<<(N_NODES * EMBED + 255) / 256, 256, 0, stream>>>(emb, x_a, acc);

  float* xs[2] = { x_a, x_b };
  const int pb = (N_NODES + WAVES - 1) / WAVES;
  for (int l = 0; l < N_LAYERS; ++l) {
    k_prop<<<pb, TPB, 0, stream>>>((const float2*)xs[0], (float2*)xs[1], (float2*)acc,
                                   offs, csr);
    float* t = xs[0]; xs[0] = xs[1]; xs[1] = t;
  }
}